// _DiagS4DLayer_84361747628471
// MI455X (gfx1250) — compile-verified
//
#include <hip/hip_runtime.h>
#include <hip/hip_bf16.h>
#include <math.h>

// Problem constants (from reference)
#define B_SZ    8
#define L_SEQ   4096
#define H_DIM   128
#define N_ST    64
#define T_CH    64                 // chunk length
#define N_CHUNK (L_SEQ / T_CH)     // 64
#define NCOL    (B_SZ * N_CHUNK)   // 512 columns for the Toeplitz GEMM
#define LN_EPS  1e-5f

typedef __attribute__((ext_vector_type(2))) float v2f;
typedef __attribute__((ext_vector_type(8))) float v8f;

// ---------------------------------------------------------------------------
// Kernel 0: build tables into workspace.
//   Ktab [H][T]        : K[h][l] = Re(sum_n CB * dA^l)
//   CAre/CAim [H][T][N]: C[h,n] * dA[h,n]^l        (inter-chunk output)
//   BPre/BPim [H][T][N]: Bbar[h,n] * dA^(T-1-t)    (state accumulation)
//   dATre/dATim [H][N] : dA^T                      (state carry)
// ---------------------------------------------------------------------------
__global__ __launch_bounds__(128) void s4d_build_tables(
    const float* __restrict__ log_neg_real, const float* __restrict__ imag,
    const float* __restrict__ B_re, const float* __restrict__ B_im,
    const float* __restrict__ C_re, const float* __restrict__ C_im,
    const float* __restrict__ log_step,
    float* __restrict__ Ktab,
    float* __restrict__ CAre, float* __restrict__ CAim,
    float* __restrict__ BPre, float* __restrict__ BPim,
    float* __restrict__ dATre, float* __restrict__ dATim)
{
    int idx = blockIdx.x * blockDim.x + threadIdx.x;   // h*T + l
    if (idx >= H_DIM * T_CH) return;
    int h = idx / T_CH;
    int l = idx % T_CH;

    float step = expf(log_step[h]);
    step = fminf(fmaxf(step, 1e-5f), 1.0f);

    float ksum = 0.0f;
    for (int n = 0; n < N_ST; ++n) {
        float lre = -expf(log_neg_real[n]);
        float sre = step * lre;
        float sim = step * imag[n];
        float cre = C_re[h * N_ST + n];
        float cim = C_im[h * N_ST + n];
        float bre = B_re[h * N_ST + n] * step;
        float bim = B_im[h * N_ST + n] * step;

        // dA^l
        float ml = expf(sre * (float)l);
        float cl = cosf(sim * (float)l);
        float sl = sinf(sim * (float)l);
        float dre = ml * cl, dim_ = ml * sl;

        // C * dA^l
        CAre[(h * T_CH + l) * N_ST + n] = cre * dre - cim * dim_;
        CAim[(h * T_CH + l) * N_ST + n] = cre * dim_ + cim * dre;

        // K[h][l] = Re(CB * dA^l), CB = C * Bbar
        float cb_re = cre * bre - cim * bim;
        float cb_im = cre * bim + cim * bre;
        ksum += cb_re * dre - cb_im * dim_;

        // Bbar * dA^(T-1-l)
        float p  = (float)(T_CH - 1 - l);
        float mp = expf(sre * p);
        float cp = cosf(sim * p);
        float sp = sinf(sim * p);
        float pre = mp * cp, pim = mp * sp;
        BPre[(h * T_CH + l) * N_ST + n] = bre * pre - bim * pim;
        BPim[(h * T_CH + l) * N_ST + n] = bre * pim + bim * pre;

        if (l == 0) {
            float mT = expf(sre * (float)T_CH);
            float cT = cosf(sim * (float)T_CH);
            float sT = sinf(sim * (float)T_CH);
            dATre[h * N_ST + n] = mT * cT;
            dATim[h * N_ST + n] = mT * sT;
        }
    }
    Ktab[idx] = ksum;
}

// ---------------------------------------------------------------------------
// Kernel A: intra-chunk causal convolution as Toeplitz GEMM with fp32 WMMA.
// For each h: Y(64 x 512) = M_h(64x64, lower-tri Toeplitz of K) @ U(64 x 512),
// column = (b, chunk).  Branchless A fragments via zero-padded Kpad[128] in
// LDS:  A[m][k] = Kpad[64 + m - k]  (Kpad[64+d] = K[h][d] for d>=0, else 0).
// Each wave owns one 16-row M tile and TWO 16-col N tiles (2 accumulators,
// A fragment reused) -> pure ds_load + v_wmma inner loop, no exec edits.
// grid = (NCOL/32, H), block = 128.  Writes y = y_intra + D*u ('=', full cover)
// ---------------------------------------------------------------------------
__global__ __launch_bounds__(128) void s4d_toeplitz_wmma(
    const float* __restrict__ u, const float* __restrict__ Ktab,
    const float* __restrict__ D, float* __restrict__ y)
{
    const int h    = blockIdx.y;
    const int col0 = blockIdx.x * 32;          // 32 columns, same b for all
    const int tid  = threadIdx.x;
    const int wave = tid >> 5;
    const int lane = tid & 31;

    __shared__ float Kpad[2 * T_CH];           // [64+d]: K[h][d] (d>=0) else 0
    __shared__ float Us[T_CH][33];             // [k][cc], padded

    Kpad[tid] = (tid >= T_CH) ? Ktab[h * T_CH + (tid - T_CH)] : 0.0f;

    const int b     = col0 >> 6;               // batch index (32 | 64)
    const int cbase = col0 & 63;               // chunk base within batch

    for (int e = tid; e < T_CH * 32; e += 128) {
        int k  = e >> 5;
        int cc = e & 31;
        int l  = (cbase + cc) * T_CH + k;
        const float* p = u + ((size_t)b * L_SEQ + l) * H_DIM + h;
        Us[k][cc] = *p;
        __builtin_prefetch(p + 1, 0, 1);       // global_prefetch_b8 (next h)
    }
    __syncthreads();

    // fp32 WMMA 16x16x4 fragments (ISA 7.12.2 layouts)
    const int half = lane >> 4;                // 0: K pair {0,1}; 1: {2,3}
    const int lid  = lane & 15;
    const int row0 = wave * 16;
    const int m    = row0 + lid;

    v8f acc0 = {};
    v8f acc1 = {};
#pragma unroll
    for (int kk = 0; kk < T_CH; kk += 4) {
        const int k0   = kk + half * 2;
        const int base = T_CH + m - k0;        // in [2,127]
        v2f a, b0, b1;
        a.x  = Kpad[base];                     // A[m][k0]
        a.y  = Kpad[base - 1];                 // A[m][k0+1]
        b0.x = Us[k0    ][lid];
        b0.y = Us[k0 + 1][lid];
        b1.x = Us[k0    ][lid + 16];
        b1.y = Us[k0 + 1][lid + 16];
        acc0 = __builtin_amdgcn_wmma_f32_16x16x4_f32(
                   false, a, false, b0, (short)0, acc0, false, false);
        acc1 = __builtin_amdgcn_wmma_f32_16x16x4_f32(
                   false, a, false, b1, (short)0, acc1, false, false);
    }

    const float dh = D[h];
#pragma unroll
    for (int r = 0; r < 8; ++r) {
        int i   = row0 + r + half * 8;         // within-chunk time index
        int cc0 = lid;
        int cc1 = lid + 16;
        int l0  = (cbase + cc0) * T_CH + i;
        int l1  = (cbase + cc1) * T_CH + i;
        y[((size_t)b * L_SEQ + l0) * H_DIM + h] = acc0[r] + dh * Us[i][cc0];
        y[((size_t)b * L_SEQ + l1) * H_DIM + h] = acc1[r] + dh * Us[i][cc1];
    }
}

// ---------------------------------------------------------------------------
// Kernel B: sequential inter-chunk state scan.  One WG per (b,h), 64 threads
// own 64 complex states.  Adds Re(C * dA^l * x_chunk_start) into y.
// ---------------------------------------------------------------------------
__global__ __launch_bounds__(64) void s4d_state_scan(
    const float* __restrict__ u,
    const float* __restrict__ CAre, const float* __restrict__ CAim,
    const float* __restrict__ BPre, const float* __restrict__ BPim,
    const float* __restrict__ dATre, const float* __restrict__ dATim,
    float* __restrict__ y)
{
    const int bh  = blockIdx.x;
    const int b   = bh / H_DIM;
    const int h   = bh % H_DIM;
    const int tid = threadIdx.x;               // 0..63: state n AND output l

    __shared__ float sCAr[T_CH * N_ST];        // 16 KB
    __shared__ float sCAi[T_CH * N_ST];        // 16 KB
    __shared__ float xre[N_ST], xim[N_ST], uc[T_CH];

    for (int e = tid; e < T_CH * N_ST; e += 64) {
        sCAr[e] = CAre[(size_t)h * T_CH * N_ST + e];
        sCAi[e] = CAim[(size_t)h * T_CH * N_ST + e];
    }
    const float dre  = dATre[h * N_ST + tid];
    const float dim_ = dATim[h * N_ST + tid];
    xre[tid] = 0.0f;
    xim[tid] = 0.0f;
    __syncthreads();

    for (int c = 0; c < N_CHUNK; ++c) {
        uc[tid] = u[((size_t)b * L_SEQ + c * T_CH + tid) * H_DIM + h];
        __syncthreads();

        // output contribution at l = tid from chunk-start state
        if (c > 0) {
            float acc = 0.0f;
#pragma unroll 8
            for (int n = 0; n < N_ST; ++n)
                acc += sCAr[tid * N_ST + n] * xre[n]
                     - sCAi[tid * N_ST + n] * xim[n];
            y[((size_t)b * L_SEQ + c * T_CH + tid) * H_DIM + h] += acc;
        }

        // state update for n = tid: x = dA^T * x + sum_t BP[t][n] * u[t]
        float nre = dre * xre[tid] - dim_ * xim[tid];
        float nim = dre * xim[tid] + dim_ * xre[tid];
#pragma unroll 8
        for (int t = 0; t < T_CH; ++t) {
            float uv = uc[t];
            nre += BPre[((size_t)h * T_CH + t) * N_ST + tid] * uv;  // coalesced, L2-resident
            nim += BPim[((size_t)h * T_CH + t) * N_ST + tid] * uv;
        }
        __syncthreads();
        xre[tid] = nre;
        xim[tid] = nim;
        __syncthreads();
    }
}

// ---------------------------------------------------------------------------
// Kernel C: exact GELU + LayerNorm over H, in place.  grid = B*L, block = 128.
// ---------------------------------------------------------------------------
__global__ __launch_bounds__(128) void s4d_gelu_ln(
    float* __restrict__ y, const float* __restrict__ gamma,
    const float* __restrict__ beta)
{
    const int row = blockIdx.x;                // b*L + l
    const int tid = threadIdx.x;               // 0..127 = channel h
    const size_t idx = (size_t)row * H_DIM + tid;

    float v = y[idx];
    float g = 0.5f * v * (1.0f + erff(v * 0.70710678118654752f));

    float s = g, s2 = g * g;
    for (int off = 16; off > 0; off >>= 1) {   // wave32 reduction
        s  += __shfl_down(s,  off, 32);
        s2 += __shfl_down(s2, off, 32);
    }
    __shared__ float ws[4], ws2[4];
    const int wave = tid >> 5, lane = tid & 31;
    if (lane == 0) { ws[wave] = s; ws2[wave] = s2; }
    __syncthreads();

    float tot  = ws[0] + ws[1] + ws[2] + ws[3];
    float tot2 = ws2[0] + ws2[1] + ws2[2] + ws2[3];
    float mean = tot / (float)H_DIM;
    float var  = tot2 / (float)H_DIM - mean * mean;
    y[idx] = (g - mean) * rsqrtf(var + LN_EPS) * gamma[tid] + beta[tid];
}

// ---------------------------------------------------------------------------
extern "C" void kernel_launch(void* const* d_in, const int* in_sizes, int n_in,
                              void* d_out, int out_size, void* d_ws, size_t ws_size,
                              hipStream_t stream) {
    const float* u            = (const float*)d_in[0];
    const float* log_neg_real = (const float*)d_in[1];
    const float* imag_        = (const float*)d_in[2];
    const float* B_re         = (const float*)d_in[3];
    const float* B_im         = (const float*)d_in[4];
    const float* C_re         = (const float*)d_in[5];
    const float* C_im         = (const float*)d_in[6];
    const float* D            = (const float*)d_in[7];
    const float* log_step     = (const float*)d_in[8];
    const float* gamma        = (const float*)d_in[9];
    const float* beta         = (const float*)d_in[10];
    float* y = (float*)d_out;

    // workspace partition (floats)
    float* ws = (float*)d_ws;
    const size_t szKT  = (size_t)H_DIM * T_CH;           // 8192
    const size_t szHTN = (size_t)H_DIM * T_CH * N_ST;    // 524288
    const size_t szHN  = (size_t)H_DIM * N_ST;           // 8192
    float* Ktab  = ws;                 ws += szKT;
    float* CAre  = ws;                 ws += szHTN;
    float* CAim  = ws;                 ws += szHTN;
    float* BPre  = ws;                 ws += szHTN;
    float* BPim  = ws;                 ws += szHTN;
    float* dATre = ws;                 ws += szHN;
    float* dATim = ws;                 ws += szHN;
    (void)ws_size; (void)in_sizes; (void)n_in; (void)out_size;

    // 1) tables
    s4d_build_tables<<<(H_DIM * T_CH + 127) / 128, 128, 0, stream>>>(
        log_neg_real, imag_, B_re, B_im, C_re, C_im, log_step,
        Ktab, CAre, CAim, BPre, BPim, dATre, dATim);

    // 2) intra-chunk Toeplitz GEMM (WMMA) + skip, writes y
    dim3 gA(NCOL / 32, H_DIM);
    s4d_toeplitz_wmma<<<gA, 128, 0, stream>>>(u, Ktab, D, y);

    // 3) inter-chunk state scan, adds into y
    s4d_state_scan<<<B_SZ * H_DIM, 64, 0, stream>>>(
        u, CAre, CAim, BPre, BPim, dATre, dATim, y);

    // 4) GELU + LayerNorm in place
    s4d_gelu_ln<<<B_SZ * L_SEQ, 128, 0, stream>>>(y, gamma, beta);
}